// GCN_84267258347718
// MI455X (gfx1250) — compile-verified
//
#include <hip/hip_runtime.h>

// ---------------------------------------------------------------------------
// GCN forward: out = spmm(relu(spmm(x@W1+b1)) @ W2 + b2)
// GEMMs use native f32 WMMA (V_WMMA_F32_16X16X4_F32) with:
//   - W staged TRANSPOSED in LDS so each B fragment is one ds_load_b64
//   - 2x row blocking per wave (each B fragment feeds two WMMAs)
// SpMM uses an on-device CSR so both aggregation layers are L2-resident
// gathers (192MB L2 holds the 51MB/25MB support matrices) with no output
// atomics.
// ---------------------------------------------------------------------------

typedef __attribute__((ext_vector_type(2))) float v2f;
typedef __attribute__((ext_vector_type(8))) float v8f;

// ---------------------------- CSR construction -----------------------------

__global__ __launch_bounds__(256) void zero_ints(int* __restrict__ p, int n) {
    int i = blockIdx.x * 256 + threadIdx.x;
    if (i < n) p[i] = 0;
}

__global__ __launch_bounds__(256) void count_rows(const int* __restrict__ row,
                                                  int* __restrict__ counts,
                                                  int E, int N) {
    int e = blockIdx.x * 256 + threadIdx.x;
    if (e >= E) return;
    int r = row[e];
    if ((unsigned)r < (unsigned)N) atomicAdd(&counts[r], 1);
}

// Exclusive scan, 1024 elements per 256-thread block (4 per thread).
__global__ __launch_bounds__(256) void scan_block(int* __restrict__ data,
                                                  int* __restrict__ bsums, int n) {
    __shared__ int sh[256];
    const int tid  = threadIdx.x;
    const int base = blockIdx.x * 1024 + tid * 4;
    int x0 = (base + 0 < n) ? data[base + 0] : 0;
    int x1 = (base + 1 < n) ? data[base + 1] : 0;
    int x2 = (base + 2 < n) ? data[base + 2] : 0;
    int x3 = (base + 3 < n) ? data[base + 3] : 0;
    int tot = x0 + x1 + x2 + x3;
    sh[tid] = tot;
    __syncthreads();
    for (int off = 1; off < 256; off <<= 1) {
        int t = (tid >= off) ? sh[tid - off] : 0;
        __syncthreads();
        sh[tid] += t;
        __syncthreads();
    }
    int ex = sh[tid] - tot;  // exclusive prefix of this thread's 4 items
    if (base + 0 < n) data[base + 0] = ex;
    if (base + 1 < n) data[base + 1] = ex + x0;
    if (base + 2 < n) data[base + 2] = ex + x0 + x1;
    if (base + 3 < n) data[base + 3] = ex + x0 + x1 + x2;
    if (tid == 255) bsums[blockIdx.x] = sh[255];
}

// Exclusive scan of <=256 block sums in a single block.
__global__ __launch_bounds__(256) void scan_sums(int* __restrict__ bsums, int nb) {
    __shared__ int sh[256];
    const int tid = threadIdx.x;
    int v = (tid < nb) ? bsums[tid] : 0;
    sh[tid] = v;
    __syncthreads();
    for (int off = 1; off < 256; off <<= 1) {
        int t = (tid >= off) ? sh[tid - off] : 0;
        __syncthreads();
        sh[tid] += t;
        __syncthreads();
    }
    if (tid < nb) bsums[tid] = sh[tid] - v;
}

// Add block offsets; also mirror rowptr[0..N) into the fill cursor.
__global__ __launch_bounds__(256) void scan_add(int* __restrict__ data,
                                                const int* __restrict__ bsums,
                                                int* __restrict__ cursor,
                                                int n, int ncur) {
    const int tid  = threadIdx.x;
    const int base = blockIdx.x * 1024 + tid * 4;
    const int add  = bsums[blockIdx.x];
#pragma unroll
    for (int j = 0; j < 4; ++j) {
        int i = base + j;
        if (i < n) {
            int v = data[i] + add;
            data[i] = v;
            if (i < ncur) cursor[i] = v;
        }
    }
}

__global__ __launch_bounds__(256) void fill_csr(const int* __restrict__ row,
                                                const int* __restrict__ col,
                                                const float* __restrict__ val,
                                                int* __restrict__ cursor,
                                                int* __restrict__ cols_s,
                                                float* __restrict__ vals_s,
                                                int E, int N) {
    int e = blockIdx.x * 256 + threadIdx.x;
    if (e >= E) return;
    int r = row[e];
    if ((unsigned)r >= (unsigned)N) return;
    int p = atomicAdd(&cursor[r], 1);
    cols_s[p] = col[e];
    vals_s[p] = val[e];
}

// ------------------------- WMMA f32 GEMM + bias ----------------------------
// Block = 256 threads (8 wave32). Block computes a 256 x NCOL output slab.
// Wave w owns two 16-row stripes (w*16 and 128+w*16) and all NCOL/16 column
// tiles, so each B fragment load feeds two WMMAs.
//
// LDS layouts (all +4-float padded rows, stride 36 -> bank-conflict-free):
//   As[r][k]   row-major A chunk       (256 x 32)
//   Bt[n][k]   TRANSPOSED W chunk      (NCOL x 32): B fragment = ds_load_b64
//
// Fragment layouts per ISA 7.12.2 (f32, 16x16x4):
//   A (16x4):  lane<16: v0=A[m][0], v1=A[m][1]; lane>=16: v0=A[m][2], v1=A[m][3]
//   B (4x16):  lane<16: v0=B[0][n], v1=B[1][n]; lane>=16: v0=B[2][n], v1=B[3][n]
//   C/D(16x16): vgpr g: lane<16 -> (M=g, N=lane); lane>=16 -> (M=g+8, N=lane-16)

template <int KDIM, int NCOL, bool RELU_IN>
__global__ __launch_bounds__(256) void gemm_bias_wmma(const float* __restrict__ A,
                                                      const float* __restrict__ W,
                                                      const float* __restrict__ bias,
                                                      float* __restrict__ out,
                                                      int nrows) {
    constexpr int KC = 32;         // K-chunk staged in LDS
    constexpr int AP = KC + 4;     // padded row stride (floats) = 36
    constexpr int NT = NCOL / 16;  // column tiles per wave
    constexpr int ROWS = 256;      // rows per block

    __shared__ float As[ROWS * AP];
    __shared__ float Bt[NCOL * AP];   // Bt[n*AP + k] = W[k0+k][n]

    const int tid  = threadIdx.x;
    const int wave = tid >> 5;
    const int lane = tid & 31;
    const int m    = lane & 15;         // fragment row / col within tile
    const int koff = (lane >> 4) * 2;   // K sub-offset per lane half
    const int blockRow = blockIdx.x * ROWS;

    v8f acc[2][NT];
#pragma unroll
    for (int s = 0; s < 2; ++s)
#pragma unroll
        for (int t = 0; t < NT; ++t)
#pragma unroll
            for (int g = 0; g < 8; ++g) acc[s][t][g] = 0.0f;

    for (int k0 = 0; k0 < KDIM; k0 += KC) {
        // ---- stage A chunk: 256 rows x 32 floats (2048 float4, 8 per thread)
#pragma unroll
        for (int i = 0; i < 8; ++i) {
            int idx = tid + i * 256;
            int r   = idx >> 3;   // 0..255
            int q   = idx & 7;    // float4 within 32-float chunk
            int gr  = blockRow + r;
            float4 v = make_float4(0.f, 0.f, 0.f, 0.f);
            if (gr < nrows)
                v = *reinterpret_cast<const float4*>(A + (size_t)gr * KDIM + k0 + q * 4);
            if (RELU_IN) {
                v.x = fmaxf(v.x, 0.f); v.y = fmaxf(v.y, 0.f);
                v.z = fmaxf(v.z, 0.f); v.w = fmaxf(v.w, 0.f);
            }
            *reinterpret_cast<float4*>(&As[r * AP + q * 4]) = v;  // AP*4B = 144B, 16B aligned
        }
        // ---- stage W chunk transposed: Bt[n][k] (coalesced global float4 reads)
        constexpr int BQ = NCOL / 4;          // float4 per W row
#pragma unroll
        for (int i = 0; i < (KC * BQ) / 256; ++i) {
            int idx = tid + i * 256;
            int kr  = idx / BQ;
            int c   = (idx % BQ) * 4;
            float4 v = *reinterpret_cast<const float4*>(W + (size_t)(k0 + kr) * NCOL + c);
            Bt[(c + 0) * AP + kr] = v.x;
            Bt[(c + 1) * AP + kr] = v.y;
            Bt[(c + 2) * AP + kr] = v.z;
            Bt[(c + 3) * AP + kr] = v.w;
        }
        __syncthreads();

        const float* a0p = &As[(wave * 16 + m) * AP + koff];
        const float* a1p = a0p + 128 * AP;
        const float* btp = &Bt[m * AP + koff];
#pragma unroll
        for (int k = 0; k < KC; k += 4) {
            v2f a0 = *reinterpret_cast<const v2f*>(a0p + k);
            v2f a1 = *reinterpret_cast<const v2f*>(a1p + k);
#pragma unroll
            for (int t = 0; t < NT; ++t) {
                v2f b = *reinterpret_cast<const v2f*>(btp + t * 16 * AP + k);
                acc[0][t] = __builtin_amdgcn_wmma_f32_16x16x4_f32(
                    false, a0, false, b, (short)0, acc[0][t], false, false);
                acc[1][t] = __builtin_amdgcn_wmma_f32_16x16x4_f32(
                    false, a1, false, b, (short)0, acc[1][t], false, false);
            }
        }
        __syncthreads();
    }

    // ---- epilogue: + bias, store
#pragma unroll
    for (int s = 0; s < 2; ++s) {
        const int rbase = blockRow + s * 128 + wave * 16 + (lane >> 4) * 8;
#pragma unroll
        for (int t = 0; t < NT; ++t) {
            float bv = bias[t * 16 + m];
#pragma unroll
            for (int g = 0; g < 8; ++g) {
                int gr = rbase + g;
                if (gr < nrows) out[(size_t)gr * NCOL + t * 16 + m] = acc[s][t][g] + bv;
            }
        }
    }
}

// ------------------------------ CSR SpMM -----------------------------------
// One wave per output row; lanes cover the D features with float4/float2.
// `dense` (support matrix) fits in the 192MB L2, so gathers are L2 hits.

template <int D>
__global__ __launch_bounds__(256) void spmm_csr(const int* __restrict__ rowptr,
                                                const int* __restrict__ cols,
                                                const float* __restrict__ vals,
                                                const float* __restrict__ dense,
                                                float* __restrict__ out,
                                                int nrows) {
    const int lane = threadIdx.x & 31;
    const int row  = blockIdx.x * 8 + (threadIdx.x >> 5);
    if (row >= nrows) return;
    constexpr int VPL = D / 32;  // floats per lane (4 for D=128, 2 for D=64)

    float acc[VPL];
#pragma unroll
    for (int v = 0; v < VPL; ++v) acc[v] = 0.0f;

    const int e0 = rowptr[row];
    const int e1 = rowptr[row + 1];
    for (int e = e0; e < e1; ++e) {
        const int   c = cols[e];
        const float w = vals[e];
        const float* p = dense + (size_t)c * D + lane * VPL;
        if (VPL == 4) {
            float4 d = *reinterpret_cast<const float4*>(p);
            acc[0] += w * d.x; acc[1] += w * d.y;
            acc[2] += w * d.z; acc[3] += w * d.w;
        } else {
            float2 d = *reinterpret_cast<const float2*>(p);
            acc[0] += w * d.x; acc[1] += w * d.y;
        }
    }
    float* po = out + (size_t)row * D + lane * VPL;
    if (VPL == 4) {
        float4 r; r.x = acc[0]; r.y = acc[1]; r.z = acc[2]; r.w = acc[3];
        *reinterpret_cast<float4*>(po) = r;
    } else {
        float2 r; r.x = acc[0]; r.y = acc[1];
        *reinterpret_cast<float2*>(po) = r;
    }
}

// ------------------------------- launcher ----------------------------------

extern "C" void kernel_launch(void* const* d_in, const int* in_sizes, int n_in,
                              void* d_out, int out_size, void* d_ws, size_t ws_size,
                              hipStream_t stream) {
    constexpr int NFEAT = 512, NHID = 128, NCLASS = 64;
    const int N = in_sizes[0] / NFEAT;   // 100000
    const int E = in_sizes[1];           // 1600000

    const float* x        = (const float*)d_in[0];
    const int*   edge_row = (const int*)d_in[1];
    const int*   edge_col = (const int*)d_in[2];
    const float* edge_val = (const float*)d_in[3];
    const float* W1       = (const float*)d_in[4];
    const float* b1       = (const float*)d_in[5];
    const float* W2       = (const float*)d_in[6];
    const float* b2       = (const float*)d_in[7];
    float*       out      = (float*)d_out;

    // Workspace carve-out (256B aligned slices).
    char*  ws  = (char*)d_ws;
    size_t off = 0;
    auto carve = [&](size_t bytes) -> void* {
        void* p = ws + off;
        off += (bytes + 255) & ~(size_t)255;
        return p;
    };
    float* support = (float*)carve((size_t)N * NHID * sizeof(float)); // support1, then support2
    float* h       = (float*)carve((size_t)N * NHID * sizeof(float)); // spmm1 output
    int*   rowptr  = (int*)carve((size_t)(N + 1) * sizeof(int));
    int*   cursor  = (int*)carve((size_t)N * sizeof(int));
    int*   bsums   = (int*)carve(256 * sizeof(int));
    int*   cols_s  = (int*)carve((size_t)E * sizeof(int));
    float* vals_s  = (float*)carve((size_t)E * sizeof(float));
    (void)ws_size; (void)n_in; (void)out_size;

    const int nscan = N + 1;
    const int nb1   = (nscan + 1023) / 1024;   // 98 blocks (<=256 required)

    // 1) CSR build (reused by both aggregation layers).
    zero_ints<<<(nscan + 255) / 256, 256, 0, stream>>>(rowptr, nscan);
    count_rows<<<(E + 255) / 256, 256, 0, stream>>>(edge_row, rowptr, E, N);
    scan_block<<<nb1, 256, 0, stream>>>(rowptr, bsums, nscan);
    scan_sums<<<1, 256, 0, stream>>>(bsums, nb1);
    scan_add<<<nb1, 256, 0, stream>>>(rowptr, bsums, cursor, nscan, N);
    fill_csr<<<(E + 255) / 256, 256, 0, stream>>>(edge_row, edge_col, edge_val,
                                                  cursor, cols_s, vals_s, E, N);

    const int gemm_blocks = (N + 255) / 256;
    const int spmm_blocks = (N + 7) / 8;

    // 2) Layer 1: support1 = x @ W1 + b1  (f32 WMMA), then aggregate.
    gemm_bias_wmma<NFEAT, NHID, false><<<gemm_blocks, 256, 0, stream>>>(x, W1, b1, support, N);
    spmm_csr<NHID><<<spmm_blocks, 256, 0, stream>>>(rowptr, cols_s, vals_s, support, h, N);

    // 3) Layer 2: support2 = relu(h) @ W2 + b2 (relu fused into A load), aggregate to out.
    gemm_bias_wmma<NHID, NCLASS, true><<<gemm_blocks, 256, 0, stream>>>(h, W2, b2, support, N);
    spmm_csr<NCLASS><<<spmm_blocks, 256, 0, stream>>>(rowptr, cols_s, vals_s, support, out, N);
}